// GAT_41618233098833
// MI455X (gfx1250) — compile-verified
//
#include <hip/hip_runtime.h>
#include <math.h>

// Problem constants (match reference)
#define B_  8
#define N_  10000
#define E_  320000
#define F_  128
#define C_  64
#define EN_ (E_ + N_)        // 330000 edges incl. self loops
#define BN_ (B_ * N_)        // 80000 node-batch rows

typedef float v2f __attribute__((ext_vector_type(2)));
typedef float v8f __attribute__((ext_vector_type(8)));

// ---------------------------------------------------------------------------
// out[b,n,c] = bias[c]   (scatter-add accumulates on top)
__global__ void k_init_out(float* __restrict__ out, const float* __restrict__ bias) {
    int i = blockIdx.x * 256 + threadIdx.x;      // grid sized exactly
    out[i] = bias[i & (C_ - 1)];
}

// ---------------------------------------------------------------------------
// watt = sum_c W_edge[0,c] * att_edge[c]   (single wave32)
__global__ void k_watt(const float* __restrict__ W_edge,
                       const float* __restrict__ att_edge,
                       float* __restrict__ watt) {
    int l = threadIdx.x;                                  // 32 lanes
    float p = W_edge[l] * att_edge[l] + W_edge[l + 32] * att_edge[l + 32];
#pragma unroll
    for (int off = 16; off > 0; off >>= 1) p += __shfl_down(p, off, 32);
    if (l == 0) *watt = p;
}

// ---------------------------------------------------------------------------
// sum(edge_attr) -> scal[0]  (mean computed as sum * 1/E at use site)
__global__ void k_easum(const float* __restrict__ edge_attr, float* __restrict__ sum) {
    __shared__ float red[256];
    float p = 0.f;
    for (int i = blockIdx.x * 256 + threadIdx.x; i < E_; i += gridDim.x * 256)
        p += edge_attr[i];
    red[threadIdx.x] = p;
    __syncthreads();
    for (int s = 128; s > 0; s >>= 1) {
        if (threadIdx.x < s) red[threadIdx.x] += red[threadIdx.x + s];
        __syncthreads();
    }
    if (threadIdx.x == 0) atomicAdd(sum, red[0]);
}

// ---------------------------------------------------------------------------
// h[b,n,c] = sum_f data[b,n,f] * W[f,c]   via V_WMMA_F32_16X16X4_F32
// Block = 256 threads = 8 waves arranged 2(M) x 4(N): 32x64 tile per block.
// A(16x4 f32): lane holds row m=lane&15; VGPR pair = K {2*(lane>>4), +1}
// B(4x16 f32): lane holds col n=lane&15; VGPR pair = K {2*(lane>>4), +1}
// C/D(16x16 f32): vgpr r -> M = r + 8*(lane>>4), N = lane&15
__global__ void __launch_bounds__(256)
k_gemm_wmma(const float* __restrict__ data, const float* __restrict__ W,
            float* __restrict__ h) {
    __shared__ float sW[F_ * C_];                 // 32 KB: whole W tile
    int t = threadIdx.x;
    const float4* Wv  = reinterpret_cast<const float4*>(W);
    float4*       sWv = reinterpret_cast<float4*>(sW);
#pragma unroll
    for (int i = 0; i < 8; ++i) sWv[t + i * 256] = Wv[t + i * 256];
    __syncthreads();

    int wave = t >> 5, lane = t & 31;
    int wm = wave >> 2, wn = wave & 3;            // 2 x 4 wave layout
    int half = lane >> 4, mloc = lane & 15;
    int row = blockIdx.x * 32 + wm * 16 + mloc;   // A row (M)
    int col = wn * 16 + mloc;                     // B/C col (N)
    const float* Arow = data + (size_t)row * F_;

    v8f acc = {0.f, 0.f, 0.f, 0.f, 0.f, 0.f, 0.f, 0.f};
#pragma unroll 8
    for (int k = 0; k < F_; k += 4) {
        int kb = k + 2 * half;
        v2f a = *reinterpret_cast<const v2f*>(Arow + kb);   // 8B aligned
        v2f b;
        b.x = sW[kb * C_ + col];
        b.y = sW[(kb + 1) * C_ + col];
        acc = __builtin_amdgcn_wmma_f32_16x16x4_f32(
                  false, a, false, b, (short)0, acc, false, false);
    }

    int rbase = blockIdx.x * 32 + wm * 16 + half * 8;
#pragma unroll
    for (int r = 0; r < 8; ++r)
        h[(size_t)(rbase + r) * C_ + col] = acc[r];
}

// ---------------------------------------------------------------------------
// a_src[b,n] = h[b,n,:] . att_src ; a_dst likewise
__global__ void k_attn(const float* __restrict__ h,
                       const float* __restrict__ att_src,
                       const float* __restrict__ att_dst,
                       float* __restrict__ a_src, float* __restrict__ a_dst) {
    int i = blockIdx.x * 256 + threadIdx.x;
    if (i >= BN_) return;
    const float4* hr = reinterpret_cast<const float4*>(h + (size_t)i * C_);
    const float4* as = reinterpret_cast<const float4*>(att_src);
    const float4* ad = reinterpret_cast<const float4*>(att_dst);
    float s = 0.f, d = 0.f;
#pragma unroll
    for (int j = 0; j < 16; ++j) {
        float4 v = hr[j], a = as[j], b = ad[j];
        s += v.x * a.x + v.y * a.y + v.z * a.z + v.w * a.w;
        d += v.x * b.x + v.y * b.y + v.z * b.z + v.w * b.w;
    }
    a_src[i] = s;
    a_dst[i] = d;
}

// Order-preserving float<->uint for atomic max
__device__ __forceinline__ unsigned fenc(float f) {
    unsigned u = __float_as_uint(f);
    return (u & 0x80000000u) ? ~u : (u | 0x80000000u);
}
__device__ __forceinline__ float fdec(unsigned u) {
    return __uint_as_float((u & 0x80000000u) ? (u ^ 0x80000000u) : ~u);
}

// ---------------------------------------------------------------------------
// Pass 1: logits -> alpha slot, segment max via atomic umax
__global__ void k_logits(const int* __restrict__ eidx,
                         const float* __restrict__ edge_attr,
                         const float* __restrict__ a_src,
                         const float* __restrict__ a_dst,
                         const float* __restrict__ scal,   // [sum_ea, watt]
                         float* __restrict__ alpha,
                         unsigned* __restrict__ useg) {
    int e = blockIdx.x * 256 + threadIdx.x;
    if (e >= EN_) return;
    int b = blockIdx.y;
    int s, d; float ea;
    if (e < E_) { s = eidx[e]; d = eidx[E_ + e]; ea = edge_attr[e]; }
    else        { s = d = e - E_; ea = scal[0] * (1.0f / E_); }   // mean attr
    float l = a_src[b * N_ + s] + a_dst[b * N_ + d] + ea * scal[1];
    l = (l > 0.f) ? l : 0.2f * l;                                  // leaky relu
    alpha[(size_t)b * EN_ + e] = l;
    atomicMax(&useg[b * N_ + d], fenc(l));
}

// ---------------------------------------------------------------------------
// Pass 2: ex = exp(l - segmax) -> alpha slot; denom += ex
__global__ void k_exp(const int* __restrict__ eidx,
                      float* __restrict__ alpha,
                      const unsigned* __restrict__ useg,
                      float* __restrict__ denom) {
    int e = blockIdx.x * 256 + threadIdx.x;
    if (e >= EN_) return;
    int b = blockIdx.y;
    int d = (e < E_) ? eidx[E_ + e] : (e - E_);
    size_t ai = (size_t)b * EN_ + e;
    float ex = expf(alpha[ai] - fdec(useg[b * N_ + d]));
    alpha[ai] = ex;
    atomicAdd(&denom[b * N_ + d], ex);
}

// ---------------------------------------------------------------------------
// Pass 3: alpha = ex/denom; out[b,dst,:] += alpha * h[b,src,:]
// One wave32 per (b,e); lane covers 2 channels (float2) -> 256B/wave gather.
__global__ void __launch_bounds__(256)
k_scatter(const int* __restrict__ eidx,
          float* __restrict__ alpha,
          const float* __restrict__ denom,
          const float* __restrict__ h,
          float* __restrict__ out) {
    int lane = threadIdx.x & 31;
    int e = blockIdx.x * 8 + (threadIdx.x >> 5);
    if (e >= EN_) return;
    int b = blockIdx.y;
    int s, d;
    if (e < E_) { s = eidx[e]; d = eidx[E_ + e]; }
    else        { s = d = e - E_; }
    size_t ai = (size_t)b * EN_ + e;
    float a = alpha[ai] / (denom[b * N_ + d] + 1e-16f);
    if (lane == 0) alpha[ai] = a;                 // final alpha output
    const float2 hv =
        reinterpret_cast<const float2*>(h + (size_t)(b * N_ + s) * C_)[lane];
    float* orow = out + (size_t)(b * N_ + d) * C_ + 2 * lane;
    atomicAdd(orow,     a * hv.x);
    atomicAdd(orow + 1, a * hv.y);
}

// ---------------------------------------------------------------------------
extern "C" void kernel_launch(void* const* d_in, const int* in_sizes, int n_in,
                              void* d_out, int out_size, void* d_ws, size_t ws_size,
                              hipStream_t stream) {
    const float* data      = (const float*)d_in[0];
    const int*   eidx      = (const int*)  d_in[1];
    const float* edge_attr = (const float*)d_in[2];
    const float* W         = (const float*)d_in[3];
    const float* att_src   = (const float*)d_in[4];
    const float* att_dst   = (const float*)d_in[5];
    const float* W_edge    = (const float*)d_in[6];
    const float* att_edge  = (const float*)d_in[7];
    const float* bias      = (const float*)d_in[8];

    float* out   = (float*)d_out;                       // [B,N,C]
    float* alpha = out + (size_t)BN_ * C_;              // [B,E+N]

    // workspace layout (floats)
    float*    ws    = (float*)d_ws;
    float*    h     = ws;                               // BN_*C_
    float*    a_src = ws + (size_t)BN_ * C_;            // BN_
    float*    a_dst = a_src + BN_;                      // BN_
    unsigned* useg  = (unsigned*)(a_dst + BN_);         // BN_ (atomic max)
    float*    denom = (float*)(useg + BN_);             // BN_ (atomic add)
    float*    scal  = denom + BN_;                      // [sum_ea, watt]

    // zero the atomically-accumulated regions (contiguous): useg,denom,scal
    hipMemsetAsync(useg, 0, (size_t)(2 * BN_ + 2) * sizeof(float), stream);

    k_init_out<<<(BN_ * C_) / 256, 256, 0, stream>>>(out, bias);
    k_watt<<<1, 32, 0, stream>>>(W_edge, att_edge, scal + 1);
    k_easum<<<128, 256, 0, stream>>>(edge_attr, scal);
    k_gemm_wmma<<<BN_ / 32, 256, 0, stream>>>(data, W, h);
    k_attn<<<(BN_ + 255) / 256, 256, 0, stream>>>(h, att_src, att_dst, a_src, a_dst);

    dim3 ge((EN_ + 255) / 256, B_);
    k_logits<<<ge, 256, 0, stream>>>(eidx, edge_attr, a_src, a_dst, scal, alpha, useg);
    k_exp<<<ge, 256, 0, stream>>>(eidx, alpha, useg, denom);

    dim3 gs((EN_ + 7) / 8, B_);
    k_scatter<<<gs, 256, 0, stream>>>(eidx, alpha, denom, h, out);
}